// RoiPoolingLayer_22462678958491
// MI455X (gfx1250) — compile-verified
//
#include <hip/hip_runtime.h>
#include <cstdint>

namespace {
constexpr int   kPool      = 7;
constexpr int   kHW        = 128;   // feature map is 128x128
constexpr int   kCH        = 512;   // channels (innermost, contiguous)
constexpr float kInvStride = 1.0f / 16.0f;

typedef float v4f __attribute__((ext_vector_type(4)));  // native vector: OK for NT builtins

struct Axis { int i0; int i1; float frac; };

// Exact mirror of reference _axis_coords for a single pool index p.
__device__ __forceinline__ Axis axis_coords(float start, float size, int p) {
    float g  = ((float)p + 0.5f) / (float)kPool;   // (arange+0.5)/pool
    float s  = g * size - 0.5f;
    s        = fminf(fmaxf(s, 0.0f), size - 1.0f);
    float f0 = floorf(s);
    float fr = s - f0;
    float f1 = fminf(f0 + 1.0f, size - 1.0f);
    Axis a;
    a.i0 = (int)fminf(fmaxf(start + f0, 0.0f), (float)(kHW - 1));
    a.i1 = (int)fminf(fmaxf(start + f1, 0.0f), (float)(kHW - 1));
    a.frac = fr;
    return a;
}
} // namespace

// One block = one output pixel (n, py, px). 128 threads x 4 channels = 512.
// The 4 bilinear corner rows (4 x 2KB) are staged via the CDNA5 async
// global->LDS DMA path, then blended and streamed out non-temporally.
__global__ void __launch_bounds__(128)
RoiPoolingLayer_22462678958491_kernel(const float* __restrict__ image,
                                      const float* __restrict__ roi,
                                      float* __restrict__ out) {
    __shared__ __align__(16) float tile[4 * kCH];

    const int pix = blockIdx.x;                  // n*49 + py*7 + px
    const int n   = pix / (kPool * kPool);
    const int rem = pix - n * (kPool * kPool);
    const int py  = rem / kPool;
    const int px  = rem - py * kPool;
    const int t   = threadIdx.x;                 // 0..127

    const float cx = roi[n * 4 + 0];
    const float cy = roi[n * 4 + 1];
    const float w  = roi[n * 4 + 2];
    const float h  = roi[n * 4 + 3];

    // jnp.round == round-half-even == rintf (RNE mode); *(1/16) exact == /16
    const float r0 = rintf((cx - 0.5f * w) * kInvStride);
    const float c0 = rintf((cy - 0.5f * h) * kInvStride);
    const float wq = fmaxf(rintf(w * kInvStride), 1.0f);
    const float hq = fmaxf(rintf(h * kInvStride), 1.0f);

    const Axis Y = axis_coords(c0, hq, py);      // rows  (H axis)
    const Axis X = axis_coords(r0, wq, px);      // cols  (W axis)

    const float ly = Y.frac, lx = X.frac;
    const float w00 = (1.0f - ly) * (1.0f - lx);
    const float w01 = (1.0f - ly) * lx;
    const float w10 = ly * (1.0f - lx);
    const float w11 = ly * lx;

    // Each lane copies its own 16B slice of each 2KB corner row.
    const float* q00 = image + ((size_t)(Y.i0 * kHW + X.i0)) * kCH + t * 4;
    const float* q01 = image + ((size_t)(Y.i0 * kHW + X.i1)) * kCH + t * 4;
    const float* q10 = image + ((size_t)(Y.i1 * kHW + X.i0)) * kCH + t * 4;
    const float* q11 = image + ((size_t)(Y.i1 * kHW + X.i1)) * kCH + t * 4;

    // Low 32 bits of a flat pointer into LDS == DS byte offset (aperture rule).
    const uint32_t lds0 = (uint32_t)(uintptr_t)&tile[0 * kCH + t * 4];
    const uint32_t lds1 = (uint32_t)(uintptr_t)&tile[1 * kCH + t * 4];
    const uint32_t lds2 = (uint32_t)(uintptr_t)&tile[2 * kCH + t * 4];
    const uint32_t lds3 = (uint32_t)(uintptr_t)&tile[3 * kCH + t * 4];

    // CDNA5 async copy engine: global -> LDS, tracked by ASYNCcnt.
    asm volatile("global_load_async_to_lds_b128 %0, %1, off"
                 :: "v"(lds0), "v"((uint64_t)(uintptr_t)q00) : "memory");
    asm volatile("global_load_async_to_lds_b128 %0, %1, off"
                 :: "v"(lds1), "v"((uint64_t)(uintptr_t)q01) : "memory");
    asm volatile("global_load_async_to_lds_b128 %0, %1, off"
                 :: "v"(lds2), "v"((uint64_t)(uintptr_t)q10) : "memory");
    asm volatile("global_load_async_to_lds_b128 %0, %1, off"
                 :: "v"(lds3), "v"((uint64_t)(uintptr_t)q11) : "memory");

    // Each lane reads back only what its own lane's DMA wrote -> wave-local
    // wait is sufficient, no workgroup barrier required.
    asm volatile("s_wait_asynccnt 0" ::: "memory");

    const v4f a = *reinterpret_cast<const v4f*>(&tile[0 * kCH + t * 4]);
    const v4f b = *reinterpret_cast<const v4f*>(&tile[1 * kCH + t * 4]);
    const v4f c = *reinterpret_cast<const v4f*>(&tile[2 * kCH + t * 4]);
    const v4f d = *reinterpret_cast<const v4f*>(&tile[3 * kCH + t * 4]);

    const v4f o = w00 * a + w01 * b + w10 * c + w11 * d;

    // 100 MB output stream: bypass L2 retention so the 32 MB image stays hot.
    v4f* op = reinterpret_cast<v4f*>(out + (size_t)pix * kCH + t * 4);
    __builtin_nontemporal_store(o, op);
}

extern "C" void kernel_launch(void* const* d_in, const int* in_sizes, int n_in,
                              void* d_out, int out_size, void* d_ws, size_t ws_size,
                              hipStream_t stream) {
    const float* image = (const float*)d_in[0];   // (1,128,128,512) f32
    const float* roi   = (const float*)d_in[1];   // (N,4) f32
    float* out = (float*)d_out;                   // (1,N,7,7,512) f32

    const int N      = in_sizes[1] / 4;           // 1000
    const int blocks = N * kPool * kPool;         // 49,000
    RoiPoolingLayer_22462678958491_kernel<<<blocks, 128, 0, stream>>>(image, roi, out);
}